// MolGAT_54150947668225
// MI455X (gfx1250) — compile-verified
//
#include <hip/hip_runtime.h>
#include <hip/hip_bf16.h>
#include <math.h>

// ---------------------------------------------------------------------------
// MolGAT for MI455X (gfx1250): WMMA f32 16x16x4 row-panel GEMMs (A-fragment
// reused across 4-8 column tiles per wave), wave32 one-wave-per-edge
// scatter-softmax, f32 L2 atomics for all segment ops.
// ---------------------------------------------------------------------------

typedef __attribute__((ext_vector_type(2))) float v2f;
typedef __attribute__((ext_vector_type(8))) float v8f;

#define NNODES 100000
#define NEDGES 1600000
#define NGRAPH 256

// ---- float atomic max via monotonic int bit-trick (init must be -inf) -----
__device__ __forceinline__ void atomicMaxFloat(float* addr, float val) {
    int iv = __float_as_int(val);
    if (iv >= 0) atomicMax((int*)addr, iv);
    else         atomicMin((unsigned int*)addr, (unsigned int)iv);
}

// ---------------------------------------------------------------------------
// C[M x CT*16] = A[M x K] @ W[K x CT*16] + bias, optional ReLU.
// One wave computes a full 16-row x CT*16-col panel: the 16x4 A fragment is
// loaded once per K-step and fed to CT independent accumulators (back-to-back
// v_wmma_f32_16x16x4_f32, no D->A/B hazard).
// A-frag: lane (k>=2?16:0)+m holds K=base,base+1 (contiguous float2, row-major A)
// B-frag: lane holds column n = lane%16; VGPR0/1 = W[k][n], W[k+1][n]
// D:      VGPR v -> row v + 8*(lane>=16), col lane%16
// ---------------------------------------------------------------------------
template <int CT>
__global__ __launch_bounds__(32)
void wmma_gemm_panel(const float* __restrict__ A, const float* __restrict__ W,
                     const float* __restrict__ bias, float* __restrict__ C,
                     int K, int relu) {
    constexpr int Ncols = CT * 16;
    const int lane = threadIdx.x & 31;
    const int half = lane >> 4;          // 0: K pair {0,1}, 1: K pair {2,3}
    const int lm   = lane & 15;
    const int row0 = blockIdx.x * 16;
    const int m    = row0 + lm;

    v8f acc[CT];
#pragma unroll
    for (int t = 0; t < CT; ++t) acc[t] = (v8f){};

    for (int kk = 0; kk < K; kk += 4) {
        const int ka = kk + 2 * half;
        const v2f a = *(const v2f*)(A + (size_t)m * K + ka);   // 8B aligned
        const float* w0 = W + (size_t)ka * Ncols + lm;
#pragma unroll
        for (int t = 0; t < CT; ++t) {
            v2f b;
            b.x = w0[t * 16];
            b.y = w0[t * 16 + Ncols];
            acc[t] = __builtin_amdgcn_wmma_f32_16x16x4_f32(
                         false, a, false, b, (short)0, acc[t], false, false);
        }
    }
#pragma unroll
    for (int t = 0; t < CT; ++t) {
        const int col = t * 16 + lm;
        const float bv = bias[col];
#pragma unroll
        for (int v = 0; v < 8; ++v) {
            float val = acc[t][v] + bv;
            if (relu) val = fmaxf(val, 0.0f);
            C[(size_t)(row0 + v + 8 * half) * Ncols + col] = val;
        }
    }
}

// ---------------------------------------------------------------------------
__global__ void fill_kernel(float* __restrict__ p, float val, int n) {
    int i = blockIdx.x * blockDim.x + threadIdx.x;
    if (i < n) p[i] = val;
}

// ---- pass 1: e = att . leakyrelu(xl[src] + xr[dst]); atomic max per dst ---
__global__ __launch_bounds__(256)
void edge_score(const float* __restrict__ xl, const float* __restrict__ xr,
                const float* __restrict__ att,
                const int* __restrict__ srcArr, const int* __restrict__ dstArr,
                float* __restrict__ ew, float* __restrict__ nodeMax, int Etot) {
    const int lane = threadIdx.x & 31;
    const int e = blockIdx.x * (blockDim.x >> 5) + (threadIdx.x >> 5);
    if (e >= Etot) return;
    int s, d;
    if (e < NEDGES) { s = srcArr[e]; d = dstArr[e]; } else { s = d = e - NEDGES; }

    const float2 l = ((const float2*)(xl + (size_t)s * 64))[lane];
    const float2 r = ((const float2*)(xr + (size_t)d * 64))[lane];
    const float2 a = ((const float2*)att)[lane];
    float v0 = l.x + r.x; v0 = (v0 >= 0.f) ? v0 : 0.2f * v0;
    float v1 = l.y + r.y; v1 = (v1 >= 0.f) ? v1 : 0.2f * v1;
    float p = a.x * v0 + a.y * v1;
#pragma unroll
    for (int off = 16; off > 0; off >>= 1) p += __shfl_xor(p, off, 32);
    if (lane == 0) {
        ew[e] = p;
        atomicMaxFloat(nodeMax + d, p);
    }
}

// ---- pass 2: w = exp(e - m[dst]); denom[dst] += w --------------------------
__global__ void edge_exp(float* __restrict__ ew, const float* __restrict__ nodeMax,
                         float* __restrict__ denom, const int* __restrict__ dstArr,
                         int Etot) {
    int e = blockIdx.x * blockDim.x + threadIdx.x;
    if (e >= Etot) return;
    int d = (e < NEDGES) ? dstArr[e] : (e - NEDGES);
    float w = __expf(ew[e] - nodeMax[d]);
    ew[e] = w;
    atomicAdd(denom + d, w);
}

// ---- pass 3: out[dst] += (w/denom[dst]) * xl[src] --------------------------
__global__ __launch_bounds__(256)
void edge_aggr(const float* __restrict__ xl, const float* __restrict__ ew,
               const float* __restrict__ denom,
               const int* __restrict__ srcArr, const int* __restrict__ dstArr,
               float* __restrict__ out, int Etot) {
    const int lane = threadIdx.x & 31;
    const int e = blockIdx.x * (blockDim.x >> 5) + (threadIdx.x >> 5);
    if (e >= Etot) return;
    int s, d;
    if (e < NEDGES) { s = srcArr[e]; d = dstArr[e]; } else { s = d = e - NEDGES; }
    const float alpha = ew[e] / denom[d];
    const float2 l = ((const float2*)(xl + (size_t)s * 64))[lane];
    float* op = out + (size_t)d * 64 + 2 * lane;
    atomicAdd(op,     alpha * l.x);
    atomicAdd(op + 1, alpha * l.y);
}

// ---- h += bias (broadcast over rows), optional ReLU ------------------------
__global__ void bias_act(float* __restrict__ h, const float* __restrict__ bias,
                         int n, int Cmask, int relu) {
    int i = blockIdx.x * blockDim.x + threadIdx.x;
    if (i >= n) return;
    float v = h[i] + bias[i & Cmask];
    if (relu) v = fmaxf(v, 0.0f);
    h[i] = v;
}

// ---- per-column sum & sumsq (register partials, then one atomic/block) -----
__global__ void col_stats(const float* __restrict__ h, float* __restrict__ stats,
                          int rows, int C) {
    const int c = threadIdx.x;              // blockDim.x == C
    float s = 0.f, ss = 0.f;
    for (int r = blockIdx.x; r < rows; r += gridDim.x) {
        float v = h[(size_t)r * C + c];
        s += v; ss += v * v;
    }
    atomicAdd(stats + c, s);
    atomicAdd(stats + C + c, ss);
}

// ---- batchnorm (training stats) + ReLU, in place ---------------------------
__global__ void bn_relu(float* __restrict__ h, const float* __restrict__ stats,
                        const float* __restrict__ g, const float* __restrict__ be,
                        int n, int C) {
    int i = blockIdx.x * blockDim.x + threadIdx.x;
    if (i >= n) return;
    const int c = i & (C - 1);
    const float inv = 1.0f / (float)NNODES;
    const float mean = stats[c] * inv;
    const float var  = stats[C + c] * inv - mean * mean;
    float v = (h[i] - mean) * rsqrtf(var + 1e-5f) * g[c] + be[c];
    h[i] = fmaxf(v, 0.0f);
}

// ---- final 64 -> 1 projection ---------------------------------------------
__global__ void rowdot(const float* __restrict__ h, const float* __restrict__ w,
                       const float* __restrict__ b, float* __restrict__ out, int rows) {
    int i = blockIdx.x * blockDim.x + threadIdx.x;
    if (i >= rows) return;
    const float4* p  = (const float4*)(h + (size_t)i * 64);
    const float4* wp = (const float4*)w;
    float acc = 0.f;
#pragma unroll
    for (int k = 0; k < 16; ++k) {
        float4 a = p[k], ww = wp[k];
        acc += a.x * ww.x + a.y * ww.y + a.z * ww.z + a.w * ww.w;
    }
    out[i] = acc + b[0];
}

// ---- global mean pool per graph -------------------------------------------
__global__ void pool_sum(const float* __restrict__ h, const int* __restrict__ batch,
                         float* __restrict__ sums, float* __restrict__ cnt, int rows) {
    int i = blockIdx.x * blockDim.x + threadIdx.x;
    if (i >= rows) return;
    int g = batch[i];
    atomicAdd(sums + g, h[i]);
    atomicAdd(cnt + g, 1.0f);
}

__global__ void pool_final(const float* __restrict__ sums, const float* __restrict__ cnt,
                           float* __restrict__ out) {
    int g = blockIdx.x * blockDim.x + threadIdx.x;
    if (g < NGRAPH) out[g] = sums[g] / fmaxf(cnt[g], 1.0f);
}

// ---------------------------------------------------------------------------
extern "C" void kernel_launch(void* const* d_in, const int* in_sizes, int n_in,
                              void* d_out, int out_size, void* d_ws, size_t ws_size,
                              hipStream_t stream) {
    const float* x     = (const float*)d_in[0];
    const int*   ei    = (const int*)d_in[1];
    const int*   batch = (const int*)d_in[2];
    const int E    = in_sizes[1] / 2;        // 1,600,000
    const int Etot = E + NNODES;             // + self loops
    const int* srcArr = ei;
    const int* dstArr = ei + E;

    const float *Wl[3], *bl[3], *Wr[3], *br[3], *att[3], *bo[3];
    for (int l = 0; l < 3; ++l) {
        Wl[l]  = (const float*)d_in[3 + 6 * l];
        bl[l]  = (const float*)d_in[4 + 6 * l];
        Wr[l]  = (const float*)d_in[5 + 6 * l];
        br[l]  = (const float*)d_in[6 + 6 * l];
        att[l] = (const float*)d_in[7 + 6 * l];
        bo[l]  = (const float*)d_in[8 + 6 * l];
    }
    const float* mW0 = (const float*)d_in[21]; const float* mb0 = (const float*)d_in[22];
    const float* g0  = (const float*)d_in[23]; const float* be0 = (const float*)d_in[24];
    const float* mW1 = (const float*)d_in[25]; const float* mb1 = (const float*)d_in[26];
    const float* g1  = (const float*)d_in[27]; const float* be1 = (const float*)d_in[28];
    const float* mW2 = (const float*)d_in[29]; const float* mb2 = (const float*)d_in[30];

    // ---- workspace carve-up (floats) ----
    float* ws    = (float*)d_ws;
    float* hA    = ws;                               // N*128 (ping)
    float* hB    = hA + (size_t)NNODES * 128;        // N*128 (pong)
    float* xl    = hB + (size_t)NNODES * 128;        // N*64
    float* xr    = xl + (size_t)NNODES * 64;         // N*64
    float* ew    = xr + (size_t)NNODES * 64;         // Etot
    float* nm    = ew + Etot;                        // N (max / final scalar)
    float* nd    = nm + NNODES;                      // N (softmax denom)
    float* stats = nd + NNODES;                      // 256
    float* psum  = stats + 256;                      // G sums + G counts (contig)
    float* pcnt  = psum + NGRAPH;

    const int nvec   = NNODES * 64;
    const int nbElem = (nvec + 255) / 256;
    const int nbNode = (NNODES + 255) / 256;
    const int nbEdgeW = (Etot + 7) / 8;              // 8 waves / block of 256
    const int nbEdgeT = (Etot + 255) / 256;
    const int nbPanel = NNODES / 16;                 // 6250 row panels
    const float NEG_INF = -__builtin_huge_valf();

    auto gat = [&](const float* hin, float* hout, int l, int relu) {
        wmma_gemm_panel<4><<<nbPanel, 32, 0, stream>>>(hin, Wl[l], bl[l], xl, 64, 0);
        wmma_gemm_panel<4><<<nbPanel, 32, 0, stream>>>(hin, Wr[l], br[l], xr, 64, 0);
        fill_kernel<<<nbNode, 256, 0, stream>>>(nm, NEG_INF, NNODES);
        fill_kernel<<<nbNode, 256, 0, stream>>>(nd, 0.f, NNODES);
        fill_kernel<<<nbElem, 256, 0, stream>>>(hout, 0.f, nvec);
        edge_score<<<nbEdgeW, 256, 0, stream>>>(xl, xr, att[l], srcArr, dstArr, ew, nm, Etot);
        edge_exp  <<<nbEdgeT, 256, 0, stream>>>(ew, nm, nd, dstArr, Etot);
        edge_aggr <<<nbEdgeW, 256, 0, stream>>>(xl, ew, nd, srcArr, dstArr, hout, Etot);
        bias_act  <<<nbElem, 256, 0, stream>>>(hout, bo[l], nvec, 63, relu);
    };

    // ---- GAT stack ----
    gat(x,  hA, 0, 1);
    gat(hA, hB, 1, 1);
    gat(hB, hA, 2, 0);

    // ---- MLP layer 0: (N,64) @ (64,128) + BN + ReLU ----
    wmma_gemm_panel<8><<<nbPanel, 32, 0, stream>>>(hA, mW0, mb0, hB, 64, 0);
    fill_kernel<<<1, 256, 0, stream>>>(stats, 0.f, 256);
    col_stats<<<256, 128, 0, stream>>>(hB, stats, NNODES, 128);
    bn_relu<<<(NNODES * 128 + 255) / 256, 256, 0, stream>>>(hB, stats, g0, be0, NNODES * 128, 128);

    // ---- MLP layer 1: (N,128) @ (128,64) + BN + ReLU ----
    wmma_gemm_panel<4><<<nbPanel, 32, 0, stream>>>(hB, mW1, mb1, hA, 128, 0);
    fill_kernel<<<1, 128, 0, stream>>>(stats, 0.f, 128);
    col_stats<<<256, 64, 0, stream>>>(hA, stats, NNODES, 64);
    bn_relu<<<nbElem, 256, 0, stream>>>(hA, stats, g1, be1, nvec, 64);

    // ---- MLP layer 2: (N,64) @ (64,1), then mean pool per graph ----
    rowdot<<<nbNode, 256, 0, stream>>>(hA, mW2, mb2, nm, NNODES);
    fill_kernel<<<2, 256, 0, stream>>>(psum, 0.f, 2 * NGRAPH);
    pool_sum<<<nbNode, 256, 0, stream>>>(nm, batch, psum, pcnt, NNODES);
    pool_final<<<1, 256, 0, stream>>>(psum, pcnt, (float*)d_out);
}